// SynSNN_42545946034951
// MI455X (gfx1250) — compile-verified
//
#include <hip/hip_runtime.h>
#include <hip/hip_bf16.h>

// SNN: cur1 = x@W1^T + b1 (WMMA f32 GEMM), then 100-step Synaptic recurrence
// with the per-step spk1@W2^T GEMM also on the WMMA pipe.
// Layouts per CDNA5 ISA (f32 16x16x4):
//   A (16x4):  lane L: row = L%16, holds K = 2*(L/16) + {0,1} in {v.x, v.y}
//   B (4x16):  lane L: col = L%16, holds K = 2*(L/16) + {0,1} in {v.x, v.y}
//   C/D:       vgpr r, lane L: row = r + 8*(L/16), col = L%16

typedef __attribute__((ext_vector_type(2))) float v2f;
typedef __attribute__((ext_vector_type(8))) float v8f;

#define BATCH   32768
#define DSTEPS  720
#define HIDDEN  50
#define NCLS    5
#define NSTEPS  100
#define NPAD    64   // padded N for the GEMM (4 x 16-wide fragments)

// ---------------------------------------------------------------------------
// Kernel 0: zero-padded copy of W1 -> W1pad[64][720] so the GEMM inner loop
// has no divergent guarded loads (rows 50..63 are zero).
// ---------------------------------------------------------------------------
__global__ __launch_bounds__(256) void SynSNN_padW1(
    const float* __restrict__ W1, float* __restrict__ W1pad) {
  const int idx = blockIdx.x * blockDim.x + threadIdx.x;
  if (idx < NPAD * DSTEPS) {
    const int row = idx / DSTEPS;
    W1pad[idx] = (row < HIDDEN) ? W1[idx] : 0.0f;  // same flat index for row<50
  }
}

// ---------------------------------------------------------------------------
// Kernel A: cur1[B][50] = x[B][720] @ W1[50][720]^T + b1
// One wave per 16-row tile; 4 N-fragments cover N=0..63 (cols >=50 are zero).
// Inner loop: 5 unconditional global_load_b64 + 4 WMMA per 4-K step.
// ---------------------------------------------------------------------------
__global__ __launch_bounds__(256) void SynSNN_gemm_cur1(
    const float* __restrict__ x, const float* __restrict__ W1pad,
    const float* __restrict__ b1, float* __restrict__ cur1) {
  const int tid  = blockIdx.x * blockDim.x + threadIdx.x;
  const int wave = tid >> 5;
  const int lane = threadIdx.x & 31;
  const int lrow = lane & 15;   // N (for B/D) or M (for A)
  const int half = lane >> 4;   // K-half / row-half select
  const int m0   = wave * 16;
  const size_t arow = (size_t)(m0 + lrow);

  v8f acc[4];
  #pragma unroll
  for (int nf = 0; nf < 4; ++nf)
    #pragma unroll
    for (int r = 0; r < 8; ++r) acc[nf][r] = 0.0f;

  const float* aptr = x + arow * DSTEPS + 2 * half;
  const float* bptr = W1pad + (size_t)lrow * DSTEPS + 2 * half;
  for (int k = 0; k < DSTEPS; k += 4) {
    // x is single-use: stream it (non-temporal) so L2 keeps W1pad/cur1.
    v2f a = __builtin_nontemporal_load((const v2f*)(aptr + k));
    #pragma unroll
    for (int nf = 0; nf < 4; ++nf) {
      v2f b = *(const v2f*)(bptr + (size_t)nf * 16 * DSTEPS + k);
      acc[nf] = __builtin_amdgcn_wmma_f32_16x16x4_f32(
          false, a, false, b, (short)0, acc[nf], false, false);
    }
  }

  #pragma unroll
  for (int nf = 0; nf < 4; ++nf) {
    const int n = nf * 16 + lrow;
    if (n < HIDDEN) {
      const float bv = b1[n];
      #pragma unroll
      for (int r = 0; r < 8; ++r) {
        const int row = m0 + 8 * half + r;
        cur1[(size_t)row * HIDDEN + n] = acc[nf][r] + bv;
      }
    }
  }
}

// ---------------------------------------------------------------------------
// Kernel B: 100-step recurrence. One wave per 16 batch rows.
// syn1 closed form: syn1_t = cur1 * s_t, s_t = 0.9*s_{t-1} + 1 (cur1 static).
// Lane owns 26 hidden units matching the A-fragment layout; spikes are written
// straight into the A fragments. W2^T held as 13 resident B fragments.
// ---------------------------------------------------------------------------
__global__ __launch_bounds__(256) void SynSNN_loop(
    const float* __restrict__ cur1ws, const float* __restrict__ W2,
    const float* __restrict__ b2, float* __restrict__ out) {
  const int tid  = blockIdx.x * blockDim.x + threadIdx.x;
  const int wave = tid >> 5;
  const int lane = threadIdx.x & 31;
  const int lrow = lane & 15;
  const int half = lane >> 4;
  const int m0   = wave * 16;
  const size_t arow = (size_t)(m0 + lrow);

  // B fragments of W2^T: K padded 50->52, N padded 5->16 with zeros.
  v2f bw[13];
  #pragma unroll
  for (int j = 0; j < 13; ++j) {
    const int kk = 4 * j + 2 * half;
    bw[j].x = 0.0f; bw[j].y = 0.0f;
    if (lrow < NCLS && kk < HIDDEN)
      bw[j] = *(const v2f*)(W2 + (size_t)lrow * HIDDEN + kk);  // 8B aligned
  }
  const float biasv = (lrow < NCLS) ? b2[lrow] : 0.0f;

  // Layer-1 per-lane state: 26 hidden units (kk=50,51 are zero padding).
  float cur[26], mem[26];
  #pragma unroll
  for (int j = 0; j < 13; ++j) {
    const int kk = 4 * j + 2 * half;
    float c0 = 0.0f, c1 = 0.0f;
    if (kk < HIDDEN) {  // kk even => kk<50 implies kk+1<=49 valid
      v2f t = *(const v2f*)(cur1ws + arow * HIDDEN + kk);
      c0 = t.x; c1 = t.y;
    }
    cur[2 * j] = c0; cur[2 * j + 1] = c1;
    mem[2 * j] = 0.0f; mem[2 * j + 1] = 0.0f;
  }

  float syn2[8], mem2[8];
  #pragma unroll
  for (int r = 0; r < 8; ++r) { syn2[r] = 0.0f; mem2[r] = 0.0f; }

  const size_t memrec_off = (size_t)NSTEPS * BATCH * NCLS;
  float s = 0.0f;  // running geometric-sum factor for syn1

  for (int t = 0; t < NSTEPS; ++t) {
    s = 0.9f * s + 1.0f;

    // Layer 1: membrane update; spikes land directly in A fragments.
    v2f afrag[13];
    #pragma unroll
    for (int i = 0; i < 26; ++i) {
      const float old   = mem[i];
      const float reset = (old > 1.0f) ? 1.0f : 0.0f;
      const float nm    = 0.5f * old + cur[i] * s - reset;
      mem[i] = nm;
      const float spk = (nm > 1.0f) ? 1.0f : 0.0f;
      if (i & 1) afrag[i >> 1].y = spk; else afrag[i >> 1].x = spk;
    }

    // cur2 = spk1 @ W2^T + b2 on the matrix pipe (C seeded with bias).
    v8f d;
    #pragma unroll
    for (int r = 0; r < 8; ++r) d[r] = biasv;
    #pragma unroll
    for (int j = 0; j < 13; ++j)
      d = __builtin_amdgcn_wmma_f32_16x16x4_f32(
          false, afrag[j], false, bw[j], (short)0, d, false, false);

    // Layer 2: lane owns (rows m0+8*half+0..7, class lrow); store if class<5.
    #pragma unroll
    for (int r = 0; r < 8; ++r) {
      const float old2   = mem2[r];
      const float reset2 = (old2 > 1.0f) ? 1.0f : 0.0f;
      syn2[r] = 0.9f * syn2[r] + d[r];
      const float nm2 = 0.5f * old2 + syn2[r] - reset2;
      mem2[r] = nm2;
      const float spk2 = (nm2 > 1.0f) ? 1.0f : 0.0f;
      if (lrow < NCLS) {
        const int row = m0 + 8 * half + r;
        const size_t idx = ((size_t)t * BATCH + row) * NCLS + lrow;
        // Streaming output (write-once): non-temporal stores.
        __builtin_nontemporal_store(spk2, out + idx);
        __builtin_nontemporal_store(nm2, out + memrec_off + idx);
      }
    }
  }
}

extern "C" void kernel_launch(void* const* d_in, const int* in_sizes, int n_in,
                              void* d_out, int out_size, void* d_ws, size_t ws_size,
                              hipStream_t stream) {
  const float* x  = (const float*)d_in[0];
  const float* W1 = (const float*)d_in[1];
  const float* b1 = (const float*)d_in[2];
  const float* W2 = (const float*)d_in[3];
  const float* b2 = (const float*)d_in[4];
  float* out   = (float*)d_out;
  float* cur1  = (float*)d_ws;                       // 32768*50*4   = 6.55 MB
  float* W1pad = cur1 + (size_t)BATCH * HIDDEN;      // 64*720*4     = 184 KB

  SynSNN_padW1<<<(NPAD * DSTEPS + 255) / 256, 256, 0, stream>>>(W1, W1pad);
  // 32768 rows / 16 rows-per-wave = 2048 waves; 8 waves/block -> 256 blocks.
  SynSNN_gemm_cur1<<<256, 256, 0, stream>>>(x, W1pad, b1, cur1);
  SynSNN_loop<<<256, 256, 0, stream>>>(cur1, W2, b2, out);
}